// Boundary_Model_11407433138694
// MI455X (gfx1250) — compile-verified
//
#include <hip/hip_runtime.h>

// Problem dims (fixed by the reference)
#define NX 2048
#define NY 4096
// Tile: 16 rows (x) x 128 cols (y, contiguous). Exact fit: 2048/16, 4096/128.
#define TX 16
#define TY 128
#define LROW (TX + 2)   // 18 tile rows incl. +/-1 halo
#define LCOL (TY + 8)   // 136-float LDS row stride -> 544 B, 16 B aligned
#define C0   4          // LDS column holding global j0 (left halo at col 3)

// ---- CDNA5 async global->LDS path (guarded; falls back to plain ld/st) ----
#if defined(__HIP_DEVICE_COMPILE__) && \
    __has_builtin(__builtin_amdgcn_global_load_async_to_lds_b128) && \
    __has_builtin(__builtin_amdgcn_global_load_async_to_lds_b32)
#define USE_ASYNC_LDS 1
#else
#define USE_ASYNC_LDS 0
#endif

static __device__ __forceinline__ int iclampi(int v, int lo, int hi) {
  v = v < lo ? lo : v;
  return v > hi ? hi : v;
}

static __device__ __forceinline__ void tile_ld_b128(const float* g, float* l) {
#if USE_ASYNC_LDS
  typedef int v4i __attribute__((vector_size(4 * sizeof(int))));
  __builtin_amdgcn_global_load_async_to_lds_b128(
      (__attribute__((address_space(1))) v4i*)g,
      (__attribute__((address_space(3))) v4i*)l, /*offset=*/0, /*cpol=*/0);
#else
  *(float4*)l = *(const float4*)g;
#endif
}

static __device__ __forceinline__ void tile_ld_b32(const float* g, float* l) {
#if USE_ASYNC_LDS
  __builtin_amdgcn_global_load_async_to_lds_b32(
      (__attribute__((address_space(1))) int*)g,
      (__attribute__((address_space(3))) int*)l, /*offset=*/0, /*cpol=*/0);
#else
  *l = *g;
#endif
}

static __device__ __forceinline__ void tile_ld_wait() {
#if USE_ASYNC_LDS
#if __has_builtin(__builtin_amdgcn_s_wait_asynccnt)
  __builtin_amdgcn_s_wait_asynccnt(0);
#else
  asm volatile("s_wait_asynccnt 0" ::: "memory");
#endif
#endif
}

// 3-point nonuniform-grid d1/d2 coefficients; clamped base makes the
// one-sided boundary stencils use the same read pattern (f[base..base+2]).
static __device__ __forceinline__ void stencil_coeffs(
    const float* __restrict__ g, int i, int n,
    float& a1, float& b1, float& c1, float& a2, float& b2, float& c2) {
  int base = iclampi(i, 1, n - 2) - 1;
  float g0 = g[base], g1 = g[base + 1], g2 = g[base + 2];
  float hL = g1 - g0, hR = g2 - g1, hS = hL + hR;
  float r1 = 1.0f / (hL * hS);
  float r2 = 1.0f / (hL * hR);
  float r3 = 1.0f / (hR * hS);
  if (i == 0) {                       // left one-sided
    a1 = -(2.0f * hL + hR) * r1; b1 = hS * r2; c1 = -hL * r3;
  } else if (i == n - 1) {            // right one-sided
    a1 = hR * r1; b1 = -hS * r2; c1 = (hL + 2.0f * hR) * r3;
  } else {                            // centered
    a1 = -hR * r1; b1 = (hR - hL) * r2; c1 = hL * r3;
  }
  a2 = 2.0f * r1; b2 = -2.0f * r2; c2 = 2.0f * r3;
}

__global__ __launch_bounds__(256) void burgers_rhs_kernel(
    const float* __restrict__ state,  // (2, NX, NY): u then v
    const float* __restrict__ gx,     // (NX,)
    const float* __restrict__ gy,     // (NY,)
    const float* __restrict__ mu_p,   // (1,)
    float* __restrict__ out)          // (2, NX, NY): du then dv
{
  __shared__ __align__(16) float su[LROW][LCOL];
  __shared__ __align__(16) float sv[LROW][LCOL];
  __shared__ float cx[6][TX];   // a1,b1,c1,a2,b2,c2 per tile row
  __shared__ float cy[6][TY];   // a1,b1,c1,a2,b2,c2 per tile col

  const int j0  = blockIdx.x * TY;
  const int i0  = blockIdx.y * TX;
  const int tid = threadIdx.y * TY + threadIdx.x;   // block = (128, 2)

  const float* u = state;
  const float* v = state + (size_t)NX * NY;

  // ---- 1) issue async tile loads: interior as B128, halo cols as B32 ----
  // interior chunks: 2 fields x 18 rows x 32 four-float chunks = 1152
  for (int q = tid; q < 2 * LROW * (TY / 4); q += 256) {
    int field = q / (LROW * (TY / 4));
    int rem   = q % (LROW * (TY / 4));
    int row   = rem / (TY / 4);
    int ch    = rem % (TY / 4);
    int gi = iclampi(i0 - 1 + row, 0, NX - 1);
    int gj = j0 + ch * 4;                       // always in range (full tiles)
    const float* src = (field ? v : u) + (size_t)gi * NY + gj;
    float* dst = field ? &sv[row][C0 + ch * 4] : &su[row][C0 + ch * 4];
    tile_ld_b128(src, dst);
  }
  // halo columns: 2 fields x 18 rows x 2 sides = 72
  if (tid < 2 * LROW * 2) {
    int field = tid / (LROW * 2);
    int rem   = tid % (LROW * 2);
    int row   = rem / 2;
    int side  = rem % 2;
    int gi = iclampi(i0 - 1 + row, 0, NX - 1);
    int gj = iclampi(side ? (j0 + TY) : (j0 - 1), 0, NY - 1);
    const float* src = (field ? v : u) + (size_t)gi * NY + gj;
    float* dst = field ? &sv[row][side ? (C0 + TY) : (C0 - 1)]
                       : &su[row][side ? (C0 + TY) : (C0 - 1)];
    tile_ld_b32(src, dst);
  }

  // ---- 2) overlap: per-tile derivative coefficients while DMA in flight ----
  if (tid < TX) {
    float a1, b1, c1, a2, b2, c2;
    stencil_coeffs(gx, i0 + tid, NX, a1, b1, c1, a2, b2, c2);
    cx[0][tid] = a1; cx[1][tid] = b1; cx[2][tid] = c1;
    cx[3][tid] = a2; cx[4][tid] = b2; cx[5][tid] = c2;
  } else if (tid >= 32 && tid < 32 + TY) {
    int lj = tid - 32;
    float a1, b1, c1, a2, b2, c2;
    stencil_coeffs(gy, j0 + lj, NY, a1, b1, c1, a2, b2, c2);
    cy[0][lj] = a1; cy[1][lj] = b1; cy[2][lj] = c1;
    cy[3][lj] = a2; cy[4][lj] = b2; cy[5][lj] = c2;
  }

  tile_ld_wait();      // s_wait_asynccnt 0 (this wave's DMA done)
  __syncthreads();     // all waves' tile + coeff data visible

  // ---- 3) stencil compute from LDS ----
  const float mu = mu_p[0];
  const int lj = threadIdx.x;                     // 0..127
  const int j  = j0 + lj;
  const int tc0 = iclampi(j, 1, NY - 2) - 1 - j0 + C0;  // LDS base col of y-triple
  const float a1y = cy[0][lj], b1y = cy[1][lj], c1y = cy[2][lj];
  const float a2y = cy[3][lj], b2y = cy[4][lj], c2y = cy[5][lj];

  for (int li = threadIdx.y; li < TX; li += 2) {  // 8 rows per thread
    int i = i0 + li;
    int tr0 = iclampi(i, 1, NX - 2) - 1 - i0 + 1; // LDS base row of x-triple
    float a1x = cx[0][li], b1x = cx[1][li], c1x = cx[2][li];
    float a2x = cx[3][li], b2x = cx[4][li], c2x = cx[5][li];

    float uc  = su[li + 1][lj + C0];
    float vc  = sv[li + 1][lj + C0];
    float uy0 = su[li + 1][tc0], uy1 = su[li + 1][tc0 + 1], uy2 = su[li + 1][tc0 + 2];
    float ux0 = su[tr0][lj + C0], ux1 = su[tr0 + 1][lj + C0], ux2 = su[tr0 + 2][lj + C0];
    float vy0 = sv[li + 1][tc0], vy1 = sv[li + 1][tc0 + 1], vy2 = sv[li + 1][tc0 + 2];
    float vx0 = sv[tr0][lj + C0], vx1 = sv[tr0 + 1][lj + C0], vx2 = sv[tr0 + 2][lj + C0];

    float d1yu = a1y * uy0 + b1y * uy1 + c1y * uy2;
    float d2yu = a2y * uy0 + b2y * uy1 + c2y * uy2;
    float d1xu = a1x * ux0 + b1x * ux1 + c1x * ux2;
    float d2xu = a2x * ux0 + b2x * ux1 + c2x * ux2;
    float d1yv = a1y * vy0 + b1y * vy1 + c1y * vy2;
    float d2yv = a2y * vy0 + b2y * vy1 + c2y * vy2;
    float d1xv = a1x * vx0 + b1x * vx1 + c1x * vx2;
    float d2xv = a2x * vx0 + b2x * vx1 + c2x * vx2;

    float du = mu * (d2yu + d2xu) - uc * d1xu - vc * d1yu + 0.01f;
    float dv = mu * (d2yv + d2xv) - uc * d1xv - vc * d1yv;

    if (i == 0 || j == 0 || j == NY - 1) du = 0.0f;  // du BCs
    if (i == 0 || j == 0) dv = 0.0f;                 // dv BCs

    size_t idx = (size_t)i * NY + j;
    out[idx] = du;
    out[(size_t)NX * NY + idx] = dv;
  }
}

extern "C" void kernel_launch(void* const* d_in, const int* in_sizes, int n_in,
                              void* d_out, int out_size, void* d_ws, size_t ws_size,
                              hipStream_t stream) {
  (void)in_sizes; (void)n_in; (void)out_size; (void)d_ws; (void)ws_size;
  // setup_inputs order: t, state, x, y, mu
  const float* state = (const float*)d_in[1];
  const float* gx    = (const float*)d_in[2];
  const float* gy    = (const float*)d_in[3];
  const float* mu    = (const float*)d_in[4];
  float* out = (float*)d_out;

  dim3 grid(NY / TY, NX / TX);   // (32, 128)
  dim3 block(TY, 2);             // 256 threads = 8 wave32
  burgers_rhs_kernel<<<grid, block, 0, stream>>>(state, gx, gy, mu, out);
}